// GuidedSampler_72430328479836
// MI455X (gfx1250) — compile-verified
//
#include <hip/hip_runtime.h>
#include <stdint.h>

typedef __attribute__((ext_vector_type(2))) float v2f;
typedef __attribute__((ext_vector_type(8))) float v8f;

#define B_SZ   32
#define DIM    512
#define HW     4096
#define KCODES 64
#define DQn    3
// KM = KCODES*DQn = 192 rows of the fused W matrix

// ---------------------------------------------------------------------------
// Kernel 1: per-batch GEMM kv = W(192x512) x features[b](512x4096) via
// V_WMMA_F32_16X16X4_F32, with fused squared-distance epilogue:
//   sqdist[b][k] += sum_{q,n in tile} (kv[b][3k+q][n] - query[b][q][n])^2
// Block: 128 threads (4 waves). Block tile: M=192, N=64. K chunked by 32,
// double-buffered in LDS via async global->LDS copies (ASYNCcnt).
// Wave w owns M rows [48w, 48w+48) == k codes [16w, 16w+16).
// ---------------------------------------------------------------------------
__global__ __launch_bounds__(128) void kv_gemm_dist_kernel(
    const float* __restrict__ Wm,     // [192][512]
    const float* __restrict__ feat,   // [32][512][4096]
    const float* __restrict__ query,  // [32][3][4096]
    float* __restrict__ sqdist)       // [32][64]
{
    __shared__ float Bs[2][32 * 64];  // 2 x 8 KB K-chunk double buffer

    const int b    = blockIdx.y;
    const int n0   = blockIdx.x * 64;
    const int tid  = threadIdx.x;
    const int wave = tid >> 5;
    const int lane = tid & 31;
    const int lh   = lane >> 4;       // which half-wave (K/M split)
    const int l16  = lane & 15;
    const int mw   = wave * 48;       // this wave's M base

    v8f acc[3][4];
#pragma unroll
    for (int mt = 0; mt < 3; ++mt)
#pragma unroll
        for (int nt = 0; nt < 4; ++nt)
            acc[mt][nt] = (v8f)0.0f;

    const float* fb = feat + ((size_t)b * DIM) * HW + n0;

    // Issue 4 async 16B copies per thread for one 32x64 chunk (8 KB total).
    auto issue_copy = [&](int kc, int bufi) {
#pragma unroll
        for (int i = 0; i < 4; ++i) {
            const int e = i * 512 + tid * 4;
            const int r = e >> 6;
            const int c = e & 63;
            const float* src = fb + (size_t)(kc + r) * HW + c;
            const unsigned dst = (unsigned)(uintptr_t)&Bs[bufi][r * 64 + c];
            asm volatile("global_load_async_to_lds_b128 %0, %1, off"
                         :: "v"(dst), "v"(src) : "memory");
        }
    };

    issue_copy(0, 0);

    for (int ch = 0; ch < DIM / 32; ++ch) {
        const int kc = ch * 32;
        const int cb = ch & 1;
        if (ch + 1 < DIM / 32) {
            issue_copy(kc + 32, cb ^ 1);
            // async loads complete in order: <=4 outstanding => chunk ch done
            asm volatile("s_wait_asynccnt 0x4" ::: "memory");
        } else {
            asm volatile("s_wait_asynccnt 0x0" ::: "memory");
        }
        __syncthreads();

#pragma unroll
        for (int s = 0; s < 8; ++s) {
            const int lr = 4 * s + 2 * lh;    // local K row pair for this half-wave
            v2f bf[4];
#pragma unroll
            for (int nt = 0; nt < 4; ++nt) {
                bf[nt].x = Bs[cb][(lr + 0) * 64 + nt * 16 + l16];
                bf[nt].y = Bs[cb][(lr + 1) * 64 + nt * 16 + l16];
            }
#pragma unroll
            for (int mt = 0; mt < 3; ++mt) {
                const int row = mw + mt * 16 + l16;
                const int col = kc + 4 * s + 2 * lh;
                v2f af = *(const v2f*)(Wm + (size_t)row * DIM + col);
#pragma unroll
                for (int nt = 0; nt < 4; ++nt) {
                    acc[mt][nt] = __builtin_amdgcn_wmma_f32_16x16x4_f32(
                        false, af, false, bf[nt], (short)0, acc[mt][nt],
                        false, false);
                }
            }
        }
        __syncthreads();   // all waves done reading Bs[cb] before it is refilled
    }

    // Epilogue: squared distances vs query, fused into the GEMM output tiles.
    // C/D layout: VGPR r holds row m0+r (lanes 0-15) and m0+r+8 (lanes 16-31).
    const float* qb = query + (size_t)b * DQn * HW;
#pragma unroll
    for (int mt = 0; mt < 3; ++mt) {
#pragma unroll
        for (int r = 0; r < 8; ++r) {
            const int m = mw + mt * 16 + r + 8 * lh;
            const int q = m % 3;
            const int k = m / 3;
            float sq = 0.0f;
#pragma unroll
            for (int nt = 0; nt < 4; ++nt) {
                const int n = n0 + nt * 16 + l16;
                const float d = acc[mt][nt][r] - qb[q * HW + n];
                sq = fmaf(d, d, sq);
            }
            // reduce across the 16-lane group that shares row m
#pragma unroll
            for (int off = 8; off >= 1; off >>= 1)
                sq += __shfl_xor(sq, off, 32);
            if (l16 == 0)
                atomicAdd(&sqdist[b * KCODES + k], sq);
        }
    }
}

// ---------------------------------------------------------------------------
// Kernel 0: zero the squared-distance accumulators (harness does not re-zero)
// ---------------------------------------------------------------------------
__global__ void zero_ws_kernel(float* __restrict__ p, int n) {
    const int i = blockIdx.x * blockDim.x + threadIdx.x;
    if (i < n) p[i] = 0.0f;
}

// ---------------------------------------------------------------------------
// Kernel 2: per-batch argmin over k, emit codes (as float, into d_out) and
// commit loss = sum_b sqdist[b, code_b] / (B*DQ*HW). One wave, lane == b.
// ---------------------------------------------------------------------------
__global__ void argmin_loss_kernel(const float* __restrict__ sqdist,
                                   float* __restrict__ out_codes,
                                   float* __restrict__ out_loss,
                                   int* __restrict__ codes_i)
{
    const int b = threadIdx.x;           // 0..31
    float best = sqdist[b * KCODES];
    int   bi   = 0;
    for (int k = 1; k < KCODES; ++k) {
        const float v = sqdist[b * KCODES + k];
        if (v < best) { best = v; bi = k; }
    }
    codes_i[b]   = bi;
    out_codes[b] = (float)bi;

    float s = best;
#pragma unroll
    for (int off = 16; off >= 1; off >>= 1)
        s += __shfl_xor(s, off, 32);
    if (b == 0)
        out_loss[0] = s / (float)(B_SZ * DQn * HW);
}

// ---------------------------------------------------------------------------
// Kernel 3: recompute only the selected k's rows:
//   sel[b][q][n] = sum_d W[3*code_b + q][d] * features[b][d][n]
// m is wave-uniform per block (scalar W loads), features reads coalesced in n.
// ---------------------------------------------------------------------------
__global__ __launch_bounds__(256) void select_kernel(
    const float* __restrict__ Wm,
    const float* __restrict__ feat,
    const int* __restrict__ codes_i,
    float* __restrict__ out)
{
    const int b   = blockIdx.y;
    const int idx = blockIdx.x * 256 + threadIdx.x;  // 0..12287
    const int q   = idx >> 12;
    const int n   = idx & 4095;
    const int m   = codes_i[b] * 3 + q;

    const float* wr = Wm + (size_t)m * DIM;
    const float* fc = feat + ((size_t)b * DIM) * HW + n;
    float s = 0.0f;
#pragma unroll 8
    for (int d = 0; d < DIM; ++d)
        s = fmaf(wr[d], fc[(size_t)d * HW], s);
    out[(size_t)b * DQn * HW + idx] = s;
}

// ---------------------------------------------------------------------------
extern "C" void kernel_launch(void* const* d_in, const int* in_sizes, int n_in,
                              void* d_out, int out_size, void* d_ws, size_t ws_size,
                              hipStream_t stream)
{
    const float* feat  = (const float*)d_in[0];   // (32,512,64,64)
    const float* query = (const float*)d_in[1];   // (32,3,64,64)
    const float* Wm    = (const float*)d_in[2];   // (64,3,512) == (192,512)

    float* out     = (float*)d_out;               // [sel 393216][codes 32][loss 1]
    float* sqdist  = (float*)d_ws;                // 32*64 floats
    int*   codes_i = (int*)((char*)d_ws + (size_t)B_SZ * KCODES * sizeof(float));

    zero_ws_kernel<<<dim3(8), 256, 0, stream>>>(sqdist, B_SZ * KCODES);
    kv_gemm_dist_kernel<<<dim3(HW / 64, B_SZ), 128, 0, stream>>>(Wm, feat, query, sqdist);
    argmin_loss_kernel<<<1, 32, 0, stream>>>(sqdist,
                                             out + (size_t)B_SZ * DQn * HW,
                                             out + (size_t)B_SZ * DQn * HW + B_SZ,
                                             codes_i);
    select_kernel<<<dim3(DQn * HW / 256, B_SZ), 256, 0, stream>>>(Wm, feat, codes_i, out);
}